// DAV_Block_55190329754127
// MI455X (gfx1250) — compile-verified
//
#include <hip/hip_runtime.h>
#include <hip/hip_bf16.h>

typedef __attribute__((ext_vector_type(2))) float v2f;
typedef __attribute__((ext_vector_type(4))) float v4f;
typedef __attribute__((ext_vector_type(8))) float v8f;

#define BB 2
#define CC 64
#define HH 192
#define WW 320
#define DD 32
#define OUT_TILE 14          // 16x16 input tile -> 14x14 valid box-filtered interior
#define TILES_H 14           // ceil(192/14)
#define TILES_W 23           // ceil(320/14)
#define EPSV 1e-12f
#define PIX_STRIDE 20        // 16 d-values per pass, padded to 20 (bank-conflict-free, 16B-aligned quads)

// Tridiagonal ones matrix T[i][j] = (|i-j|<=1)
__device__ __forceinline__ float tband(int i, int j) {
    return ((unsigned)(i - j + 1) <= 2u) ? 1.0f : 0.0f;
}

// Compute boxed = T * X * T for a 16x16 f32 matrix X living in LDS,
// element X[row][col] at sX[(row*16+col)*pixStride + off].
// Result returned in WMMA C/D layout (8 VGPRs: lane<16 -> M=r, lane>=16 -> M=r+8; N=lane&15).
// Interior elements (1<=M<=14, 1<=N<=14) equal the 3x3 box sum of X.
__device__ __forceinline__ v8f box16(const float* __restrict__ sX, int pixStride, int off,
                                     float* __restrict__ scratch, int lane) {
    const int half = lane >> 4;   // 0 or 1
    const int nm   = lane & 15;

    // GEMM1: V = T * X   (A = T from lane-id constants, B = X from LDS)
    v8f acc = {};
#pragma unroll
    for (int k = 0; k < 4; ++k) {
        const int K0 = 4 * k + 2 * half;
        v2f a, b;
        a.x = tband(nm, K0);
        a.y = tband(nm, K0 + 1);
        b.x = sX[(K0 * 16 + nm) * pixStride + off];
        b.y = sX[((K0 + 1) * 16 + nm) * pixStride + off];
        acc = __builtin_amdgcn_wmma_f32_16x16x4_f32(
            false, a, false, b, (short)0, acc, false, false);
    }

    // Relayout V: C/D layout -> row-major LDS scratch (16 x 17, per-wave region)
#pragma unroll
    for (int r = 0; r < 8; ++r)
        scratch[(r + 8 * half) * 17 + nm] = acc[r];
    __builtin_amdgcn_wave_barrier();   // keep LDS store->load order (in-order DS pipe per wave)

    // GEMM2: O = V * T   (A = V from scratch in A-layout, B = T constants)
    v8f o = {};
#pragma unroll
    for (int k = 0; k < 4; ++k) {
        const int K0 = 4 * k + 2 * half;
        v2f a, b;
        a.x = scratch[nm * 17 + K0];
        a.y = scratch[nm * 17 + K0 + 1];
        b.x = tband(K0, nm);
        b.y = tband(K0 + 1, nm);
        o = __builtin_amdgcn_wmma_f32_16x16x4_f32(
            false, a, false, b, (short)0, o, false, false);
    }
    return o;
}

__global__ void __launch_bounds__(256)
dav_block_kernel(const float* __restrict__ q, const float* __restrict__ wf,
                 float* __restrict__ out) {
    __shared__ __align__(16) float sNum[256 * PIX_STRIDE];  // 20 KB
    __shared__ __align__(16) float sKk [256 * PIX_STRIDE];  // 20 KB
    __shared__ float sQq[256];                              // 1 KB
    __shared__ float sScr[8 * 16 * 17];                     // 8.5 KB per-wave GEMM relayout scratch

    const int tid  = threadIdx.x;
    const int lane = tid & 31;
    const int wid  = tid >> 5;

    int bx = blockIdx.x;
    const int tw = bx % TILES_W; bx /= TILES_W;
    const int th = bx % TILES_H; bx /= TILES_H;
    const int b  = bx;

    const int gh0 = th * OUT_TILE - 1;   // global row of tile-local row 0 (halo)
    const int gw0 = tw * OUT_TILE - 1;

    const size_t HW   = (size_t)HH * WW;        // q channel stride
    const size_t HWD  = HW * DD;                // wf channel stride

    float* const myScr = sScr + wid * (16 * 17);
    const int half = lane >> 4;
    const int nm   = lane & 15;

    // Two passes over d (16 displacement slices per pass) to keep LDS ~50KB.
    for (int pass = 0; pass < 2; ++pass) {
        // ---------------- Phase A: streaming channel reduction ----------------
        const int dq   = lane & 3;    // which float4 of this pass's 16 d-values
        const int psub = lane >> 2;   // 8 concurrent pixels per wave
        for (int g = 0; g < 4; ++g) {
            const int pix = wid * 32 + g * 8 + psub;   // 0..255, unique per block
            const int row = pix >> 4, col = pix & 15;
            const int gh = gh0 + row, gw = gw0 + col;

            v4f num4 = {}; v4f kk4 = {}; float qq = 0.0f;
            if (gh >= 0 && gh < HH && gw >= 0 && gw < WW) {
                const float* qp = q  + (size_t)(b * CC) * HW + (size_t)gh * WW + gw;
                const float* wp = wf + ((size_t)(b * CC) * HW + (size_t)gh * WW + gw) * DD
                                     + pass * 16 + dq * 4;
#pragma unroll 8
                for (int c = 0; c < CC; ++c) {
                    const float qv = qp[(size_t)c * HW];
                    const v4f  wv = *(const v4f*)(wp + (size_t)c * HWD);
                    num4 += qv * wv;
                    kk4  += wv * wv;
                    qq   += qv * qv;
                }
            }
            *(v4f*)(sNum + pix * PIX_STRIDE + dq * 4) = num4;
            *(v4f*)(sKk  + pix * PIX_STRIDE + dq * 4) = kk4;
            if (dq == 0) sQq[pix] = qq;
        }
        __syncthreads();

        // ---------------- Phase B: 3x3 box via f32 WMMA + normalize ----------------
        // box of qq (per-pixel scalar) — each wave computes it (cheap, avoids a broadcast sync)
        const v8f bq = box16(sQq, 1, 0, myScr, lane);

#pragma unroll
        for (int di = 0; di < 2; ++di) {
            const int dl = wid + di * 8;          // 0..15 within this pass
            const int d  = pass * 16 + dl;        // global displacement index
            const v8f bn = box16(sNum, PIX_STRIDE, dl, myScr, lane);
            const v8f bk = box16(sKk,  PIX_STRIDE, dl, myScr, lane);

#pragma unroll
            for (int r = 0; r < 8; ++r) {
                const int M = r + 8 * half;       // tile-local row in C/D layout
                const int N = nm;                 // tile-local col
                const int gh = gh0 + M, gw = gw0 + N;
                const bool valid = (M >= 1) && (M <= OUT_TILE) &&
                                   (N >= 1) && (N <= OUT_TILE) &&
                                   (gh < HH) && (gw < WW);
                if (valid) {
                    const float nq = fmaxf(sqrtf(bq[r]), EPSV);
                    const float nk = fmaxf(sqrtf(bk[r]), EPSV);
                    out[((size_t)(b * DD + d) * HH + gh) * WW + gw] = bn[r] / (nq * nk);
                }
            }
        }
        __syncthreads();   // LDS reused by next pass's Phase A
    }
}

extern "C" void kernel_launch(void* const* d_in, const int* in_sizes, int n_in,
                              void* d_out, int out_size, void* d_ws, size_t ws_size,
                              hipStream_t stream) {
    (void)in_sizes; (void)n_in; (void)d_ws; (void)ws_size; (void)out_size;
    const float* q  = (const float*)d_in[0];
    const float* wf = (const float*)d_in[1];
    float* out = (float*)d_out;

    dim3 grid(BB * TILES_H * TILES_W);   // 2 * 14 * 23 = 644 blocks
    dim3 block(256);                     // 8 wave32s
    dav_block_kernel<<<grid, block, 0, stream>>>(q, wf, out);
}